// ScaledDotProductAttention_3951369912419
// MI455X (gfx1250) — compile-verified
//
#include <hip/hip_runtime.h>
#include <math.h>

// ---------------------------------------------------------------------------
// Scaled dot-product attention, B=8 H=8 S=2048 D=7, fp32, outputs (context, attn).
// Store-bandwidth-bound on MI455X (1.07GB attn writes / 23.3TB/s -> ~47us floor),
// but borderline issue-bound, so per-tile instruction count is minimized:
//   pass 1a: QK^T via v_wmma_f32_16x16x4_f32, row-max only
//   pass 1b: recompute scores, rowsum += v_exp(fma(x, log2e, c0))
//   pass 2 : recompute, p = v_exp(fma(x, log2e, c2)) folds max+scale+1/sum,
//            coalesced b128 NT stores via LDS, PV via 4 chained WMMAs.
// ---------------------------------------------------------------------------

typedef __attribute__((ext_vector_type(2))) float v2f;
typedef __attribute__((ext_vector_type(4))) float v4f;
typedef __attribute__((ext_vector_type(8))) float v8f;

#define S_LEN      2048
#define DHEAD      7
#define NBH        64                       // B*H
#define CTX_ELEMS  (NBH * S_LEN * DHEAD)    // context floats; attn follows
#define KSTR       10                       // K rows padded 7->10: aligned b64 B-ops,
                                            // conflict-free banks, col7 pad = K=7 zero
#define VSTR       8                        // V rows padded 7->8 (pad = 0)
#define WAVES_PER_WG 8
#define PSTR       20                       // P scratch stride: 16B-aligned float4 rows
#define KTILES     (S_LEN / 16)             // 128
#define LOG2E      1.4426950408889634f

union V8 { v8f v; float f[8]; };

__device__ __forceinline__ v8f wmma4(v2f a, v2f b, v8f c) {
  // D = A(16x4 f32) * B(4x16 f32) + C(16x16 f32)
  return __builtin_amdgcn_wmma_f32_16x16x4_f32(
      false, a, false, b, (short)0, c, false, false);
}

__global__ void __launch_bounds__(256)
sdpa_wmma_kernel(const float* __restrict__ Qg,
                 const float* __restrict__ Kg,
                 const float* __restrict__ Vg,
                 float* __restrict__ out) {
  extern __shared__ float lds[];
  float* Kl = lds;                               // [2048][10]
  float* Vl = lds + S_LEN * KSTR;                // [2048][8]
  const int tid  = threadIdx.x;
  const int wave = tid >> 5;
  const int lane = tid & 31;
  const int hi   = lane >> 4;                    // half-wave select
  const int n    = lane & 15;                    // col / M index within tile
  const int bh   = blockIdx.y;
  const int qbase = (blockIdx.x * WAVES_PER_WG + wave) * 16;

  float* Pl = lds + S_LEN * (KSTR + VSTR) + wave * (16 * PSTR);

  // ---- cooperative K/V stage into LDS (zero pads double as WMMA K-dim padding)
  const float* Kbh = Kg + (size_t)bh * S_LEN * DHEAD;
  const float* Vbh = Vg + (size_t)bh * S_LEN * DHEAD;
  for (int row = tid; row < S_LEN; row += 256) {
#pragma unroll
    for (int d = 0; d < DHEAD; ++d) {
      Kl[row * KSTR + d] = Kbh[row * DHEAD + d];
      Vl[row * VSTR + d] = Vbh[row * DHEAD + d];
    }
    Kl[row * KSTR + 7] = 0.f;
    Kl[row * KSTR + 8] = 0.f;
    Kl[row * KSTR + 9] = 0.f;
    Vl[row * VSTR + 7] = 0.f;
  }
  __syncthreads();

  // ---- Q tile -> WMMA A operands, pre-scaled by 1/sqrt(7); K padded to 8
  // A 16x4 f32: lanes 0-15 hold {K=0,K=1}, lanes 16-31 {K=2,K=3}, M = lane&15
  const float scale = 0.3779644730092272f;       // rsqrt(7)
  const float* Qrow = Qg + ((size_t)bh * S_LEN + qbase + n) * DHEAD;
  v2f a_lo, a_hi;
  a_lo.x = Qrow[hi ? 2 : 0] * scale;
  a_lo.y = Qrow[hi ? 3 : 1] * scale;
  a_hi.x = Qrow[hi ? 6 : 4] * scale;
  a_hi.y = hi ? 0.f : Qrow[5] * scale;           // K=7 pad / K=5

  // score tile: 2 chained WMMAs; B operands are aligned b64 LDS loads
  auto qk_tile = [&](int kb) {
    const float* kr = &Kl[(kb + n) * KSTR];
    v2f b_lo = *(const v2f*)(kr + 2 * hi);        // {d0,d1} | {d2,d3}
    v2f b_hi = *(const v2f*)(kr + 4 + 2 * hi);    // {d4,d5} | {d6,pad0}
    V8 c; c.v = (v8f){};
    c.v = wmma4(a_lo, b_lo, c.v);
    c.v = wmma4(a_hi, b_hi, c.v);
    return c;
  };

  // ======================= pass 1a: row max =================================
  float rm[8];
#pragma unroll
  for (int r = 0; r < 8; ++r) rm[r] = -3.0e38f;
  for (int kt = 0; kt < KTILES; ++kt) {
    V8 c = qk_tile(kt * 16);
#pragma unroll
    for (int r = 0; r < 8; ++r) rm[r] = fmaxf(rm[r], c.f[r]);
  }
#pragma unroll
  for (int r = 0; r < 8; ++r) {
#pragma unroll
    for (int off = 1; off < 16; off <<= 1)
      rm[r] = fmaxf(rm[r], __shfl_xor(rm[r], off, 16));
  }
  float c0[8];
#pragma unroll
  for (int r = 0; r < 8; ++r) c0[r] = -rm[r] * LOG2E;

  // ======================= pass 1b: row sum of exp ==========================
  float rs[8];
#pragma unroll
  for (int r = 0; r < 8; ++r) rs[r] = 0.f;
  for (int kt = 0; kt < KTILES; ++kt) {
    V8 c = qk_tile(kt * 16);
#pragma unroll
    for (int r = 0; r < 8; ++r)
      rs[r] += __builtin_amdgcn_exp2f(fmaf(c.f[r], LOG2E, c0[r]));
  }
#pragma unroll
  for (int r = 0; r < 8; ++r) {
#pragma unroll
    for (int off = 1; off < 16; off <<= 1)
      rs[r] += __shfl_xor(rs[r], off, 16);
  }
  // c2 folds max-subtract AND 1/rowsum into the exp argument: no div, no mul
  float c2[8];
#pragma unroll
  for (int r = 0; r < 8; ++r) c2[r] = c0[r] - __builtin_amdgcn_logf(rs[r]);

  // ============ pass 2: normalize, NT-store attn, PV via WMMA ===============
  float* ctxP  = out + (size_t)bh * S_LEN * DHEAD;
  float* attnP = out + (size_t)CTX_ELEMS + (size_t)bh * S_LEN * S_LEN;
  const int srow = lane >> 2;                    // coalesced-store row (0..7)
  const int sgrp = (lane & 3) * 4;               // col group of 4

  V8 ctx; ctx.v = (v8f){};
  for (int kt = 0; kt < KTILES; ++kt) {
    const int kb = kt * 16;
    V8 c = qk_tile(kb);

    // p = exp2(x*log2e + c2): single FMA + v_exp per element, fully normalized
    // C layout: VGPR r = row qbase + hi*8 + r, col kb + n
#pragma unroll
    for (int r = 0; r < 8; ++r)
      Pl[(hi * 8 + r) * PSTR + n] =
          __builtin_amdgcn_exp2f(fmaf(c.f[r], LOG2E, c2[r]));
    // same-wave LDS is in-order; wait for writes, block compiler reordering
    asm volatile("s_wait_dscnt 0" ::: "memory");

    // coalesced attn stream-out: 2 x b128 NT stores per lane (16x16 tile)
    v4f plo = *(const v4f*)&Pl[srow * PSTR + sgrp];
    v4f phi = *(const v4f*)&Pl[(srow + 8) * PSTR + sgrp];
    float* dst = attnP + (size_t)(qbase + srow) * S_LEN + kb + sgrp;
    __builtin_nontemporal_store(plo, (v4f*)dst);
    __builtin_nontemporal_store(phi, (v4f*)(dst + 8 * S_LEN));

    // ctx(16x16, cols 0..6 valid) += P(16x16) * V(16x8pad): 4 K=4 WMMAs
#pragma unroll
    for (int j = 0; j < 4; ++j) {
      v2f pa = *(const v2f*)&Pl[n * PSTR + 4 * j + 2 * hi];  // A 16x4 layout
      const int krow = kb + 4 * j + 2 * hi;
      v2f vb;
      vb.x = (n < VSTR) ? Vl[krow * VSTR + n] : 0.f;         // n==7 -> pad 0
      vb.y = (n < VSTR) ? Vl[(krow + 1) * VSTR + n] : 0.f;
      ctx.v = wmma4(pa, vb, ctx.v);
    }
  }

  // ---- context tile: VGPR r = row qbase + hi*8 + r, col n (valid d < 7)
  if (n < DHEAD) {
#pragma unroll
    for (int r = 0; r < 8; ++r)
      ctxP[(size_t)(qbase + hi * 8 + r) * DHEAD + n] = ctx.f[r];
  }
}

extern "C" void kernel_launch(void* const* d_in, const int* in_sizes, int n_in,
                              void* d_out, int out_size, void* d_ws, size_t ws_size,
                              hipStream_t stream) {
  const float* Q = (const float*)d_in[0];
  const float* K = (const float*)d_in[1];
  const float* V = (const float*)d_in[2];
  float* out = (float*)d_out;

  dim3 grid(S_LEN / (WAVES_PER_WG * 16), NBH);   // (16 q-chunks, 64 bh)
  dim3 block(256);                               // 8 waves
  size_t lds_bytes =
      (size_t)(S_LEN * (KSTR + VSTR) + WAVES_PER_WG * 16 * PSTR) * sizeof(float);
  sdpa_wmma_kernel<<<grid, block, lds_bytes, stream>>>(Q, K, V, out);
}